// PISDistBlock_79637283602679
// MI455X (gfx1250) — compile-verified
//
#include <hip/hip_runtime.h>
#include <math.h>

// Problem constants from the reference
constexpr int Bn    = 256;   // batch
constexpr int Cn    = 8;     // channels
constexpr int LEN   = 4096;  // time series length
constexpr int Kk    = 128;   // shapelet length
constexpr int DIMc  = 3;     // channel used
constexpr int PLEN  = 2152;  // END - START (START=0, END=2152)
constexpr int OUTc  = PLEN - Kk + 1; // 2025
constexpr int TB    = 256;   // threads per block (8 wave32s)
constexpr int PISPAD = 2432; // padded pis length (max index used: 2176)
constexpr int KP    = 144;   // padded K' (K + 15 -> multiple of 4)

typedef __attribute__((ext_vector_type(2))) float v2f;
typedef __attribute__((ext_vector_type(8))) float v8f;

__global__ __launch_bounds__(TB)
void pis_dist_wmma_kernel(const float* __restrict__ x,
                          const float* __restrict__ shp,
                          float* __restrict__ out)
{
    __shared__ float s_pis[PISPAD + 8];  // zero-padded channel-3 slice
    __shared__ float s_sh[Kk];           // shapelet (for scalar pre-pass)
    __shared__ float s_bsh[KP * 16];     // B: banded shapelet  B[k',n] = sh[k'-n] (band 0..127)
    __shared__ float s_b28[KP * 16];     // B: 0/1 band, width 128 (for sum win^2)
    __shared__ float s_b27[KP * 16];     // B: 0/1 band, width 127 (for ci sliding sum)
    __shared__ float s_red[2 * TB];      // block reduction scratch

    const int b   = blockIdx.x;
    const int tid = threadIdx.x;

    // ---- Stage LDS: pis slice (zero padded), shapelet, banded B matrices ----
    const float* xp = x + ((size_t)b * Cn + DIMc) * LEN; // START == 0
    for (int i = tid; i < PISPAD + 8; i += TB)
        s_pis[i] = (i < PLEN) ? xp[i] : 0.0f;
    if (tid < Kk)
        s_sh[tid] = shp[tid];
    for (int i = tid; i < KP * 16; i += TB) {
        const int kp = i >> 4;
        const int n  = i & 15;
        const int d  = kp - n;                       // shapelet shift
        const bool in128 = ((unsigned)d < 128u);
        s_bsh[i] = in128 ? shp[d] : 0.0f;
        s_b28[i] = in128 ? 1.0f : 0.0f;
        s_b27[i] = ((unsigned)d < 127u) ? 1.0f : 0.0f;
    }
    __syncthreads();

    // ---- Scalars (redundant per thread: cheap + deterministic) ----
    float shsq = 0.0f, ci_sh = 0.0f;
    for (int i = 0; i < Kk; ++i)     { float s  = s_sh[i];               shsq  += s  * s;  }
    for (int i = 0; i < Kk - 1; ++i) { float df = s_sh[i + 1] - s_sh[i]; ci_sh += df * df; }
    ci_sh += 0.1f; // + 1/NORM

    // ---- WMMA tile geometry (one 16x16 output tile per wave) ----
    const int wave  = tid >> 5;
    const int lane  = tid & 31;
    const int mrow  = lane & 15;       // A row index / B,C column index (n)
    const int hi    = lane >> 4;       // half-wave selector for K-pair layout
    const int j0    = wave << 8;       // 256 outputs per wave
    const int abase = j0 + (mrow << 4);

    v8f acc_cr = {}; // sum win*sh
    v8f acc_s2 = {}; // sum win^2
    v8f acc_ci = {}; // sliding sum of d2 (window 127)

    // Banded GEMM over shifts, K' = 144, fp32 WMMA (16x16x4)
    for (int kk = 0; kk < KP; kk += 4) {
        const int k0 = kk + (hi << 1);   // lane's B rows: k0, k0+1 ; A Ks: k0, k0+1
        // A fragments: 3 contiguous pis values (q0 == p1 shared)
        const float p0 = s_pis[abase + k0];
        const float p1 = s_pis[abase + k0 + 1];
        const float p2 = s_pis[abase + k0 + 2];
        v2f a_cr = { p0, p1 };
        v2f a_s2 = { p0 * p0, p1 * p1 };
        const float e0 = p1 - p0, e1 = p2 - p1;
        v2f a_ci = { e0 * e0, e1 * e1 };

        // B fragments: straight LDS loads from precomputed banded matrices
        const int bidx = (k0 << 4) + mrow;           // row k0, col n
        v2f b_sh = { s_bsh[bidx], s_bsh[bidx + 16] };
        v2f b_s2 = { s_b28[bidx], s_b28[bidx + 16] };
        v2f b_ci = { s_b27[bidx], s_b27[bidx + 16] };

        acc_cr = __builtin_amdgcn_wmma_f32_16x16x4_f32(false, a_cr, false, b_sh,
                                                       (short)0, acc_cr, false, false);
        acc_s2 = __builtin_amdgcn_wmma_f32_16x16x4_f32(false, a_s2, false, b_s2,
                                                       (short)0, acc_s2, false, false);
        acc_ci = __builtin_amdgcn_wmma_f32_16x16x4_f32(false, a_ci, false, b_ci,
                                                       (short)0, acc_ci, false, false);
    }

    // ---- Per-output epilogue + local soft-min partials ----
    float sum_w = 0.0f, sum_wd = 0.0f;
#pragma unroll
    for (int v = 0; v < 8; ++v) {
        const int Mr = v + (hi << 3);            // C/D row: VGPR v -> M=v (lo) / v+8 (hi)
        const int j  = j0 + (Mr << 4) + mrow;    // output index
        if (j < OUTc) {
            const float cip  = acc_ci[v] + 0.1f; // ci_pis
            const float mx   = fmaxf(cip, ci_sh);
            const float mn   = fminf(cip, ci_sh);
            const float cid  = fminf(mx / mn, 3.0f);          // MAX_CI
            const float dist = (acc_s2[v] - 2.0f * acc_cr[v] + shsq) * cid * (1.0f / 128.0f);
            const float dd   = dist * 0.01f;                  // /BNORM
            const float w    = __expf(-10.0f * dd);           // ALPHA = -10
            sum_w  += w;
            sum_wd += w * dd;
        }
    }

    // ---- Deterministic block tree reduction ----
    s_red[tid]      = sum_w;
    s_red[TB + tid] = sum_wd;
    __syncthreads();
    for (int s = TB >> 1; s > 0; s >>= 1) {
        if (tid < s) {
            s_red[tid]      += s_red[tid + s];
            s_red[TB + tid] += s_red[TB + tid + s];
        }
        __syncthreads();
    }
    if (tid == 0) {
        const float softmin = (s_red[TB] / s_red[0]) * 100.0f; // *BNORM
        out[b] = 1.0f - softmin * 1.0e5f;                      // / MAX_NORM_DIST
    }
}

extern "C" void kernel_launch(void* const* d_in, const int* in_sizes, int n_in,
                              void* d_out, int out_size, void* d_ws, size_t ws_size,
                              hipStream_t stream) {
    const float* x   = (const float*)d_in[0]; // (256, 8, 4096) f32
    const float* shp = (const float*)d_in[1]; // (1, 128) f32
    // d_in[2] = ep (unused)
    float* out = (float*)d_out;               // (256, 1) f32

    pis_dist_wmma_kernel<<<dim3(Bn), dim3(TB), 0, stream>>>(x, shp, out);
}